// BatchTopKSAE_70007966925170
// MI455X (gfx1250) — compile-verified
//
#include <hip/hip_runtime.h>
#include <hip/hip_bf16.h>

typedef float v2f __attribute__((ext_vector_type(2)));
typedef float v8f __attribute__((ext_vector_type(8)));

#define D_MODEL   2048
#define D_SAE     16384
#define BATCH_N   4096
#define K_TOP     64
#define K_AUX     32
#define TOTAL_K   (K_TOP * BATCH_N)
#define AUX_PEN   (1.0f / 32.0f)

// ---------------- utility kernels ----------------

__global__ void zero_u32_kernel(unsigned* p, int n) {
    int i = blockIdx.x * 256 + threadIdx.x;
    if (i < n) p[i] = 0u;
}

// c_enc[j] = b_enc[j] - sum_k b_dec[k] * W_enc[k][j]
__global__ __launch_bounds__(256) void cenc_kernel(const float* __restrict__ W_enc,
                                                   const float* __restrict__ b_enc,
                                                   const float* __restrict__ b_dec,
                                                   float* __restrict__ c_enc) {
    int j = blockIdx.x * 256 + threadIdx.x;
    float s = 0.f;
    for (int k = 0; k < D_MODEL; ++k)
        s += b_dec[k] * W_enc[(size_t)k * D_SAE + j];
    c_enc[j] = b_enc[j] - s;
}

// ---------------- f32 WMMA GEMM: C[M,N] = A[M,K] @ B[K,N] ----------------
// Block tile 128(M) x 64(N), 8 waves, each wave owns 32x32 = 2x2 WMMA subtiles.
// LDS double-buffered (1 barrier/stage); B stored K-pair packed (float2) so each
// B frag is one b64 and bf0/bf1 merge into a single ds_load_2addr_b64.
// EPI 0: relu(acc + bias[col])           (encoder -> acts)
// EPI 1: acc + bias[col]; fused (v-Xref)^2 sum -> accum   (decoder -> xr)
template <int EPI>
__global__ __launch_bounds__(256) void wmma_gemm_kernel(
    const float* __restrict__ A, const float* __restrict__ B,
    float* __restrict__ C, const float* __restrict__ bias,
    const float* __restrict__ Xref, float* __restrict__ accum,
    int K, int N) {
    __shared__ float As[2][128 * 36];   // [buf][row*36 + k], 32 k per stage
    __shared__ float Bs[2][16 * 160];   // [buf][(k2*80 + n)*2], float2 = (B[2k2][n], B[2k2+1][n])
    __shared__ float rbuf[256];

    const int tid  = threadIdx.x;
    const int lane = tid & 31;
    const int wave = tid >> 5;
    const int l15  = lane & 15;
    const int lh   = lane >> 4;               // 0/1 half-wave
    const int m0   = blockIdx.y * 128;
    const int n0   = blockIdx.x * 64;
    const int wr   = wave >> 1;               // 0..3  (rows wr*32 .. +31)
    const int wc   = wave & 1;                // 0..1  (cols wc*32 .. +31)

    v8f acc00 = {0.f,0.f,0.f,0.f,0.f,0.f,0.f,0.f};
    v8f acc01 = {0.f,0.f,0.f,0.f,0.f,0.f,0.f,0.f};
    v8f acc10 = {0.f,0.f,0.f,0.f,0.f,0.f,0.f,0.f};
    v8f acc11 = {0.f,0.f,0.f,0.f,0.f,0.f,0.f,0.f};

    // staging maps
    const int arow = tid >> 3;                // 0..31 (+32/+64/+96)
    const int ac   = (tid & 7) << 2;          // 0..28
    const int bk2  = tid >> 4;                // 0..15 (K-pair row)
    const int bn4  = (tid & 15) << 2;         // 0..60

    const float* Ap = A + (size_t)(m0 + arow) * K + ac;
    const float* Bp = B + (size_t)(2 * bk2) * N + n0 + bn4;

    // ---- prologue: load + store stage 0 into buffer 0 ----
    float4 na0 = *(const float4*)(Ap);
    float4 na1 = *(const float4*)(Ap + (size_t)32 * K);
    float4 na2 = *(const float4*)(Ap + (size_t)64 * K);
    float4 na3 = *(const float4*)(Ap + (size_t)96 * K);
    float4 nb0 = *(const float4*)(Bp);
    float4 nb1 = *(const float4*)(Bp + N);
    {
        *(float4*)(&As[0][(arow)      * 36 + ac]) = na0;
        *(float4*)(&As[0][(arow + 32) * 36 + ac]) = na1;
        *(float4*)(&As[0][(arow + 64) * 36 + ac]) = na2;
        *(float4*)(&As[0][(arow + 96) * 36 + ac]) = na3;
        float* bdst = &Bs[0][(bk2 * 80 + bn4) * 2];
        ((v2f*)bdst)[0] = (v2f){nb0.x, nb1.x};
        ((v2f*)bdst)[1] = (v2f){nb0.y, nb1.y};
        ((v2f*)bdst)[2] = (v2f){nb0.z, nb1.z};
        ((v2f*)bdst)[3] = (v2f){nb0.w, nb1.w};
    }

    const int nst = K >> 5;   // stages of 32 K
    for (int s = 0; s < nst; ++s) {
        const int cur = s & 1;
        const bool more = (s + 1 < nst);
        if (more) {   // issue next-stage global loads before compute (uniform branch)
            const int kc = (s + 1) << 5;
            na0 = *(const float4*)(Ap + kc);
            na1 = *(const float4*)(Ap + (size_t)32 * K + kc);
            na2 = *(const float4*)(Ap + (size_t)64 * K + kc);
            na3 = *(const float4*)(Ap + (size_t)96 * K + kc);
            nb0 = *(const float4*)(Bp + (size_t)kc * N);
            nb1 = *(const float4*)(Bp + (size_t)(kc + 1) * N);
        }
        __syncthreads();   // stage-s LDS visible; buf cur^1 free to overwrite
        const float* Ab = As[cur];
        const float* Bb = Bs[cur];
#pragma unroll
        for (int kk = 0; kk < 32; kk += 4) {
            const int lk = lh << 1;                 // lane K-pair base
            const int k2 = (kk >> 1) + lh;          // packed K-pair index
            v2f af0 = *(const v2f*)(Ab + (wr * 32 + l15) * 36 + kk + lk);
            v2f af1 = *(const v2f*)(Ab + (wr * 32 + 16 + l15) * 36 + kk + lk);
            v2f bf0 = *(const v2f*)(Bb + (k2 * 80 + wc * 32 + l15) * 2);
            v2f bf1 = *(const v2f*)(Bb + (k2 * 80 + wc * 32 + 16 + l15) * 2);
            acc00 = __builtin_amdgcn_wmma_f32_16x16x4_f32(false, af0, false, bf0,
                                                          (short)0, acc00, false, false);
            acc01 = __builtin_amdgcn_wmma_f32_16x16x4_f32(false, af0, false, bf1,
                                                          (short)0, acc01, false, false);
            acc10 = __builtin_amdgcn_wmma_f32_16x16x4_f32(false, af1, false, bf0,
                                                          (short)0, acc10, false, false);
            acc11 = __builtin_amdgcn_wmma_f32_16x16x4_f32(false, af1, false, bf1,
                                                          (short)0, acc11, false, false);
        }
        if (more) {   // store next stage into the other buffer
            const int nxt = cur ^ 1;
            *(float4*)(&As[nxt][(arow)      * 36 + ac]) = na0;
            *(float4*)(&As[nxt][(arow + 32) * 36 + ac]) = na1;
            *(float4*)(&As[nxt][(arow + 64) * 36 + ac]) = na2;
            *(float4*)(&As[nxt][(arow + 96) * 36 + ac]) = na3;
            float* bdst = &Bs[nxt][(bk2 * 80 + bn4) * 2];
            ((v2f*)bdst)[0] = (v2f){nb0.x, nb1.x};
            ((v2f*)bdst)[1] = (v2f){nb0.y, nb1.y};
            ((v2f*)bdst)[2] = (v2f){nb0.z, nb1.z};
            ((v2f*)bdst)[3] = (v2f){nb0.w, nb1.w};
        }
    }

    // epilogue: C layout -> elem r: row = r + 8*(lane>>4), col = lane&15
    const int col0 = n0 + wc * 32 + l15;
    const int col1 = col0 + 16;
    const float bv0 = bias[col0];
    const float bv1 = bias[col1];
    float lsum = 0.f;
    const int rb0 = m0 + wr * 32 + (lh << 3);
    const int rb1 = rb0 + 16;
#pragma unroll
    for (int r = 0; r < 8; ++r) {
        float v00 = acc00[r] + bv0;
        float v01 = acc01[r] + bv1;
        float v10 = acc10[r] + bv0;
        float v11 = acc11[r] + bv1;
        if (EPI == 0) {
            v00 = fmaxf(v00, 0.f); v01 = fmaxf(v01, 0.f);
            v10 = fmaxf(v10, 0.f); v11 = fmaxf(v11, 0.f);
        }
        C[(size_t)(rb0 + r) * N + col0] = v00;
        C[(size_t)(rb0 + r) * N + col1] = v01;
        C[(size_t)(rb1 + r) * N + col0] = v10;
        C[(size_t)(rb1 + r) * N + col1] = v11;
        if (EPI == 1) {
            float d0 = v00 - Xref[(size_t)(rb0 + r) * N + col0];
            float d1 = v01 - Xref[(size_t)(rb0 + r) * N + col1];
            float d2 = v10 - Xref[(size_t)(rb1 + r) * N + col0];
            float d3 = v11 - Xref[(size_t)(rb1 + r) * N + col1];
            lsum += d0 * d0 + d1 * d1 + d2 * d2 + d3 * d3;
        }
    }
    if constexpr (EPI == 1) {
        rbuf[tid] = lsum;
        __syncthreads();
        for (int s = 128; s > 0; s >>= 1) {
            if (tid < s) rbuf[tid] += rbuf[tid + s];
            __syncthreads();
        }
        if (tid == 0) atomicAdd(accum, rbuf[0]);
    }
}

// ---------------- global top-k threshold (two-level radix histogram) ----------------

__global__ void hist_hi_kernel(const float* __restrict__ acts, unsigned* __restrict__ hist) {
    const size_t nv = (size_t)BATCH_N * D_SAE / 4;
    const size_t stride = (size_t)gridDim.x * blockDim.x;
    for (size_t i = (size_t)blockIdx.x * blockDim.x + threadIdx.x; i < nv; i += stride) {
        float4 v = ((const float4*)acts)[i];
        float a[4] = {v.x, v.y, v.z, v.w};
#pragma unroll
        for (int j = 0; j < 4; ++j)
            if (a[j] > 0.f) atomicAdd(&hist[__float_as_uint(a[j]) >> 16], 1u);
    }
}

__global__ void hist_lo_kernel(const float* __restrict__ acts, const unsigned* __restrict__ scal,
                               unsigned* __restrict__ hist2) {
    const unsigned selhi = scal[0];
    const size_t nv = (size_t)BATCH_N * D_SAE / 4;
    const size_t stride = (size_t)gridDim.x * blockDim.x;
    for (size_t i = (size_t)blockIdx.x * blockDim.x + threadIdx.x; i < nv; i += stride) {
        float4 v = ((const float4*)acts)[i];
        float a[4] = {v.x, v.y, v.z, v.w};
#pragma unroll
        for (int j = 0; j < 4; ++j) {
            unsigned b = __float_as_uint(a[j]);
            if (a[j] > 0.f && (b >> 16) == selhi) atomicAdd(&hist2[b & 0xFFFFu], 1u);
        }
    }
}

// find largest bin b such that suffix count >= target; pass0 -> scal[0,1], pass1 -> scal[2,3,4]
__global__ void scan_kernel(const unsigned* __restrict__ hist, unsigned* __restrict__ scal,
                            int pass) {
    __shared__ unsigned csum[256];
    __shared__ unsigned bins[256];
    __shared__ unsigned s_selc, s_cum, s_target;
    const int tid = threadIdx.x;
    unsigned target = (pass == 0) ? (unsigned)TOTAL_K : scal[1];
    unsigned s = 0;
    for (int i = 0; i < 256; ++i) s += hist[tid * 256 + i];
    csum[tid] = s;
    __syncthreads();
    if (tid == 0) {
        unsigned cum = 0; int selc = 0;
        for (int c = 255; c >= 0; --c) {
            if (cum + csum[c] >= target) { selc = c; break; }
            cum += csum[c];
        }
        s_selc = (unsigned)selc; s_cum = cum; s_target = target;
    }
    __syncthreads();
    bins[tid] = hist[s_selc * 256 + tid];
    __syncthreads();
    if (tid == 0) {
        unsigned cum = s_cum, target2 = s_target;
        unsigned sel = s_selc * 256, need = 1;
        for (int b = 255; b >= 0; --b) {
            if (cum + bins[b] >= target2) {
                sel = s_selc * 256 + (unsigned)b;
                need = target2 - cum;
                break;
            }
            cum += bins[b];
            if (b == 0) { sel = s_selc * 256; need = (target2 > cum) ? (target2 - cum) : 1u; }
        }
        if (pass == 0) { scal[0] = sel; scal[1] = need; }
        else {
            scal[2] = (scal[0] << 16) | sel;   // exact 32-bit threshold bit pattern
            scal[3] = need;                     // number of elems == thr to keep
            scal[4] = 0;                        // tie ticket counter
        }
    }
}

// write acts_topk, accumulate L1 sum + L0 count
__global__ __launch_bounds__(256) void scatter_kernel(const float* __restrict__ acts,
                                                      float* __restrict__ topk,
                                                      unsigned* __restrict__ scal) {
    __shared__ float fbuf[256];
    __shared__ unsigned ubuf[256];
    const unsigned thr = scal[2];
    const unsigned keq = scal[3];
    float lsum = 0.f; unsigned lcnt = 0;
    const size_t nv = (size_t)BATCH_N * D_SAE / 4;
    const size_t stride = (size_t)gridDim.x * blockDim.x;
    for (size_t i = (size_t)blockIdx.x * blockDim.x + threadIdx.x; i < nv; i += stride) {
        float4 v = ((const float4*)acts)[i];
        float a[4] = {v.x, v.y, v.z, v.w};
        float o[4];
#pragma unroll
        for (int j = 0; j < 4; ++j) {
            unsigned b = __float_as_uint(a[j]);
            bool keep = false;
            if (a[j] > 0.f) {
                if (b > thr) keep = true;
                else if (b == thr) keep = (atomicAdd(&scal[4], 1u) < keq);
            }
            o[j] = keep ? a[j] : 0.f;
            lsum += o[j];
            lcnt += keep ? 1u : 0u;
        }
        float4 ov = {o[0], o[1], o[2], o[3]};
        ((float4*)topk)[i] = ov;
    }
    fbuf[threadIdx.x] = lsum; ubuf[threadIdx.x] = lcnt;
    __syncthreads();
    for (int s = 128; s > 0; s >>= 1) {
        if (threadIdx.x < s) { fbuf[threadIdx.x] += fbuf[threadIdx.x + s]; ubuf[threadIdx.x] += ubuf[threadIdx.x + s]; }
        __syncthreads();
    }
    if (threadIdx.x == 0) {
        atomicAdd((float*)scal + 5, fbuf[0]);
        atomicAdd(&scal[6], ubuf[0]);
    }
}

// ---------------- per-row top-32 over dead features ----------------
__global__ __launch_bounds__(256) void aux_topk_kernel(const float* __restrict__ acts,
                                                       const int* __restrict__ nba,
                                                       float* __restrict__ aux_vals,
                                                       int* __restrict__ aux_idx) {
    __shared__ float sv[D_SAE];
    __shared__ float lmaxs[256];
    __shared__ int   largs[256];
    __shared__ float rv[256];
    __shared__ int   rf[256];
    __shared__ int   winner;
    const int row = blockIdx.x, tid = threadIdx.x;
    const float* arow = acts + (size_t)row * D_SAE;
    float best = -3.0f; int bf = tid;
    for (int i = 0; i < D_SAE / 256; ++i) {
        int f = i * 256 + tid;
        float m = (nba[f] >= 10) ? arow[f] : -1.0f;   // sentinel like reference
        sv[f] = m;
        if (m > best || (m == best && f < bf)) { best = m; bf = f; }
    }
    lmaxs[tid] = best; largs[tid] = bf;
    __syncthreads();
    for (int it = 0; it < K_AUX; ++it) {
        rv[tid] = lmaxs[tid]; rf[tid] = largs[tid];
        __syncthreads();
        for (int s = 128; s > 0; s >>= 1) {
            if (tid < s) {
                if (rv[tid + s] > rv[tid] ||
                    (rv[tid + s] == rv[tid] && rf[tid + s] < rf[tid])) {
                    rv[tid] = rv[tid + s]; rf[tid] = rf[tid + s];
                }
            }
            __syncthreads();
        }
        if (tid == 0) {
            aux_vals[row * K_AUX + it] = fmaxf(rv[0], 0.f);   // clamp like reference
            aux_idx[row * K_AUX + it] = rf[0];
            winner = rf[0];
        }
        __syncthreads();
        int wf = winner;
        if ((wf & 255) == tid) {       // owner removes winner, rescans its stripe
            sv[wf] = -3.0f;
            float nb = -3.0f; int nf = tid;
            for (int i = 0; i < D_SAE / 256; ++i) {
                int f = i * 256 + tid;
                float m = sv[f];
                if (m > nb) { nb = m; nf = f; }
            }
            lmaxs[tid] = nb; largs[tid] = nf;
        }
        __syncthreads();
    }
}

// sparse aux reconstruction + fused (xra - residual)^2 accumulation
__global__ __launch_bounds__(256) void aux_recon_kernel(const float* __restrict__ W_dec,
                                                        const float* __restrict__ x,
                                                        const float* __restrict__ xr,
                                                        const float* __restrict__ aux_vals,
                                                        const int* __restrict__ aux_idx,
                                                        unsigned* __restrict__ scal) {
    __shared__ float buf[256];
    const int row = blockIdx.x, tid = threadIdx.x;
    float acc[D_MODEL / 256];
#pragma unroll
    for (int j = 0; j < D_MODEL / 256; ++j) acc[j] = 0.f;
    for (int s = 0; s < K_AUX; ++s) {
        float v = aux_vals[row * K_AUX + s];      // uniform per block
        if (v > 0.f) {
            const float* wrow = W_dec + (size_t)aux_idx[row * K_AUX + s] * D_MODEL;
#pragma unroll
            for (int j = 0; j < D_MODEL / 256; ++j) acc[j] += v * wrow[tid + 256 * j];
        }
    }
    float lsum = 0.f;
#pragma unroll
    for (int j = 0; j < D_MODEL / 256; ++j) {
        int c = tid + 256 * j;
        float r = x[(size_t)row * D_MODEL + c] - xr[(size_t)row * D_MODEL + c];
        float d = acc[j] - r;
        lsum += d * d;
    }
    buf[tid] = lsum;
    __syncthreads();
    for (int s = 128; s > 0; s >>= 1) {
        if (tid < s) buf[tid] += buf[tid + s];
        __syncthreads();
    }
    if (tid == 0) atomicAdd((float*)scal + 8, buf[0]);
}

__global__ __launch_bounds__(256) void finalize_kernel(const int* __restrict__ nba,
                                                       const unsigned* __restrict__ scal,
                                                       float* __restrict__ outs) {
    __shared__ unsigned cge[256], cgt[256];
    const int tid = threadIdx.x;
    unsigned ge = 0, gt = 0;
    for (int i = tid; i < D_SAE; i += 256) {
        int v = nba[i];
        ge += (v >= 10) ? 1u : 0u;
        gt += (v > 10) ? 1u : 0u;
    }
    cge[tid] = ge; cgt[tid] = gt;
    __syncthreads();
    for (int s = 128; s > 0; s >>= 1) {
        if (tid < s) { cge[tid] += cge[tid + s]; cgt[tid] += cgt[tid + s]; }
        __syncthreads();
    }
    if (tid == 0) {
        const float* fs = (const float*)scal;
        const float inv_bd = 1.0f / ((float)BATCH_N * (float)D_MODEL);
        float l2   = fs[7] * inv_bd;
        float l1n  = fs[5] / (float)BATCH_N;
        float l1l  = 0.0f;                        // L1_COEFF == 0
        float l0   = (float)scal[6] / (float)BATCH_N;
        float aux  = (cge[0] > 0) ? (AUX_PEN * fs[8] * inv_bd) : 0.0f;
        float loss = l2 + l1l + aux;
        outs[0] = loss; outs[1] = l2; outs[2] = l1l; outs[3] = l0;
        outs[4] = l1n;  outs[5] = aux; outs[6] = (float)cgt[0];
    }
}

// ---------------- launch ----------------

extern "C" void kernel_launch(void* const* d_in, const int* in_sizes, int n_in,
                              void* d_out, int out_size, void* d_ws, size_t ws_size,
                              hipStream_t stream) {
    (void)in_sizes; (void)n_in; (void)out_size; (void)ws_size;
    const float* x     = (const float*)d_in[0];
    const float* W_enc = (const float*)d_in[1];
    const float* W_dec = (const float*)d_in[2];
    const float* b_enc = (const float*)d_in[3];
    const float* b_dec = (const float*)d_in[4];
    const int*   nba   = (const int*)d_in[5];

    // workspace layout
    float*    acts     = (float*)d_ws;                          // 4096*16384
    float*    c_enc    = acts + (size_t)BATCH_N * D_SAE;        // 16384
    unsigned* hist1    = (unsigned*)(c_enc + D_SAE);            // 65536
    unsigned* hist2    = hist1 + 65536;                          // 65536
    unsigned* scal     = hist2 + 65536;                          // 64 (counters/accums)
    float*    aux_vals = (float*)(scal + 64);                    // 4096*32
    int*      aux_idx  = (int*)(aux_vals + (size_t)BATCH_N * K_AUX);

    // output layout: xr, acts_topk, then 7 scalars
    float* xr   = (float*)d_out;
    float* topk = xr + (size_t)BATCH_N * D_MODEL;
    float* outs = topk + (size_t)BATCH_N * D_SAE;

    // 0) zero histograms + scalar accumulators (every call: deterministic)
    {
        int n = 65536 * 2 + 64;
        zero_u32_kernel<<<(n + 255) / 256, 256, 0, stream>>>(hist1, n);
    }
    // 1) fused encoder bias
    cenc_kernel<<<D_SAE / 256, 256, 0, stream>>>(W_enc, b_enc, b_dec, c_enc);
    // 2) encoder GEMM (WMMA f32): acts = relu(x @ W_enc + c_enc)
    wmma_gemm_kernel<0><<<dim3(D_SAE / 64, BATCH_N / 128), 256, 0, stream>>>(
        x, W_enc, acts, c_enc, nullptr, nullptr, D_MODEL, D_SAE);
    // 3) global top-k threshold
    hist_hi_kernel<<<4096, 256, 0, stream>>>(acts, hist1);
    scan_kernel<<<1, 256, 0, stream>>>(hist1, scal, 0);
    hist_lo_kernel<<<4096, 256, 0, stream>>>(acts, scal, hist2);
    scan_kernel<<<1, 256, 0, stream>>>(hist2, scal, 1);
    // 4) scatter acts_topk + L1/L0 accumulation
    scatter_kernel<<<4096, 256, 0, stream>>>(acts, topk, scal);
    // 5) decoder GEMM (WMMA f32): xr = acts_topk @ W_dec + b_dec, fused L2 sum
    wmma_gemm_kernel<1><<<dim3(D_MODEL / 64, BATCH_N / 128), 256, 0, stream>>>(
        topk, W_dec, xr, b_dec, x, (float*)scal + 7, D_SAE, D_MODEL);
    // 6) per-row aux top-32 over dead features
    aux_topk_kernel<<<BATCH_N, 256, 0, stream>>>(acts, nba, aux_vals, aux_idx);
    // 7) sparse aux reconstruction + aux L2 accumulation
    aux_recon_kernel<<<BATCH_N, 256, 0, stream>>>(W_dec, x, xr, aux_vals, aux_idx, scal);
    // 8) scalars
    finalize_kernel<<<1, 256, 0, stream>>>(nba, scal, outs);
}